// ACELoss_14018773254365
// MI455X (gfx1250) — compile-verified
//
#include <hip/hip_runtime.h>
#include <stdint.h>

// Problem constants (from the reference): B=32 batch, D=128 feat, P=3136 patches.
#define B_  32
#define D_  128
#define P_  3136
#define COV_S 129   // padded LDS row stride for the Cov tile (TDM pads 1 dword / 128 dwords)
#define XC_S  33    // padded stride for Xc[d][b]

typedef __attribute__((ext_vector_type(2))) float        v2f;
typedef __attribute__((ext_vector_type(8))) float        v8f;
typedef __attribute__((ext_vector_type(4))) unsigned int v4u;
typedef __attribute__((ext_vector_type(8))) int          v8i;
typedef __attribute__((ext_vector_type(4))) int          v4i;

__global__ __launch_bounds__(256) void ace_kernel(
    const float* __restrict__ X,          // [B, D, P]
    const float* __restrict__ b_mean,     // [D, P]
    const float* __restrict__ b_covs,     // [P, D, D]
    const float* __restrict__ signatures, // [D, P]
    const int*   __restrict__ cov_type_p, // scalar
    float*       __restrict__ out)        // [B, P]
{
  extern __shared__ float smem[];
  float* sCov  = smem;                       // 128*129  (Cov tile, padded rows)
  float* sY    = sCov  + D_ * COV_S;         // 128*128  (solve scratch; reused as W for cov_type==0)
  float* sXc   = sY    + D_ * D_;            // 128*33   (Xc[d][b])
  float* sSig  = sXc   + D_ * XC_S;          // 128
  float* sDiag = sSig  + D_;                 // 128  (holds 1/diag(inv Cov))
  float* sColk = sDiag + D_;                 // 128  (scaled pivot column)
  float* sRed  = sColk + D_;                 // 32*8*2
  float* sNs   = sRed  + 32 * 8 * 2;         // 1

  const int tid  = threadIdx.x;
  const int lane = tid & 31;
  const int wid  = tid >> 5;
  const int p    = blockIdx.x;
  const int ctype = cov_type_p[0];

  // ---- Phase 0: TDM DMA of b_covs[p] (contiguous 64KB) into LDS, 1-dword row pad.
  if (ctype != 2 && wid == 0) {
    const uint64_t ga  = (uint64_t)(uintptr_t)(b_covs + (size_t)p * D_ * D_);
    const uint32_t lds = (uint32_t)(uintptr_t)sCov;   // low 32 bits of generic ptr == LDS byte addr
    // D# group0: count=1, lds_addr, global_addr[56:0], type=2
    v4u g0 = { 1u, lds, (uint32_t)ga,
               (((uint32_t)(ga >> 32)) & 0x01FFFFFFu) | 0x80000000u };
    // D# group1: data_size=4B(2), pad_enable, pad_interval=128dw(6), pad_amount=1dw(0),
    //            tensor_dim0=128 @bit48, tensor_dim1=128 @bit80, tile_dim0=128 @bit112,
    //            tile_dim1=128 @bit128, tensor_dim0_stride=128 @bit160
    v8i g1 = { (int)((2u << 16) | (1u << 20) | (6u << 22)),
               (int)(128u << 16),   // tensor_dim0 low16 @ [31:16]
               (int)(128u << 16),   // tensor_dim1 low16 @ [95:80]
               (int)(128u << 16),   // tile_dim0 @ [127:112]
               (int)128,            // tile_dim1 @ [143:128]
               (int)128,            // tensor_dim0_stride @ [191:160]
               0, 0 };
    v4i z4 = { 0, 0, 0, 0 };
#if defined(__clang_major__) && (__clang_major__ >= 23)
    v8i z8 = { 0, 0, 0, 0, 0, 0, 0, 0 };
    __builtin_amdgcn_tensor_load_to_lds(g0, g1, z4, z4, z8, 0);
#else
    __builtin_amdgcn_tensor_load_to_lds(g0, g1, z4, z4, 0);
#endif
  }

  // ---- Phase 1 (overlapped with DMA): stage Xc[d][b] and signatures column.
  //      Lanes walk d fastest: stride P*4 bytes in both X and b_mean.
  for (int q = tid; q < D_ * B_; q += 256) {
    const int d = q & 127;
    const int b = q >> 7;
    sXc[d * XC_S + b] = X[((size_t)b * D_ + d) * P_ + p] - b_mean[(size_t)d * P_ + p];
  }
  if (tid < D_) sSig[tid] = signatures[(size_t)tid * P_ + p];

  if (ctype != 2 && wid == 0) __builtin_amdgcn_s_wait_tensorcnt(0);
  __syncthreads();

  if (tid == 0) {
    float a = 0.f;
    for (int d = 0; d < D_; ++d) a += sSig[d] * sSig[d];
    sNs[0] = fmaxf(sqrtf(a), 1e-12f);
  }
  __syncthreads();

  // ---- Phase 2a: cov_type==1 -> 1/diag(inv(Cov)) via Cholesky + 128 forward solves.
  if (ctype == 1) {
    for (int k = 0; k < D_; ++k) {
      // Scale pivot column into sColk (sColk[k] = sqrt(a_kk) = a_kk*rsqrt(a_kk)),
      // write scaled values back into the column. 2 barriers per k-step total.
      const float linv = rsqrtf(sCov[k * COV_S + k]);
      for (int m = k + tid; m < D_; m += 256) {
        const float v = sCov[m * COV_S + k] * linv;
        sColk[m] = v;
        sCov[m * COV_S + k] = v;
      }
      __syncthreads();
      // Rank-1 update of the trailing lower triangle; two threads split each row.
      {
        const int m    = k + 1 + (tid & 127);
        const int half = tid >> 7;
        if (m < D_) {
          const float lmk = sColk[m];
          const int cnt = m - k;              // j in [k+1, m], cnt elements
          const int first = cnt >> 1;
          const int j0 = (half == 0) ? (k + 1) : (k + 1 + first);
          const int j1 = (half == 0) ? (k + 1 + first) : (m + 1);
          for (int j = j0; j < j1; ++j)
            sCov[m * COV_S + j] -= lmk * sColk[j];
        }
      }
      __syncthreads();
    }
    // Thread i solves L y = e_i ; diag(inv)_i = ||y||^2 (independent, no barriers).
    // Store the reciprocal so phase 3 multiplies instead of divides.
    if (tid < D_) {
      const int i = tid;
      float dsum = 0.f;
      for (int k = i; k < D_; ++k) {
        float acc = (k == i) ? 1.0f : 0.0f;
        for (int j = i; j < k; ++j) acc -= sCov[k * COV_S + j] * sY[j * D_ + i];
        const float yk = acc / sCov[k * COV_S + k];
        sY[k * D_ + i] = yk;
        dsum += yk * yk;
      }
      sDiag[i] = 1.0f / dsum;
    }
  }

  // ---- Phase 2b: cov_type==0 -> W[m,b] = Cov x Xc via FP32 WMMA 16x16x4.
  if (ctype == 0) {
    const int m0   = wid * 16;             // 8 waves cover M=128
    const int arow = m0 + (lane & 15);     // A: lanes 0-15 M rows, K pair by lane[4]
    const int koff = (lane >> 4) << 1;
    const int nn   = lane & 15;            // B/C/D: N striped over lanes 0-15
    v8f acc0 = {}; v8f acc1 = {};
    for (int k0 = 0; k0 < D_; k0 += 4) {
      v2f a;  a.x  = sCov[arow * COV_S + k0 + koff];
              a.y  = sCov[arow * COV_S + k0 + koff + 1];
      v2f b0; b0.x = sXc[(k0 + koff) * XC_S + nn];
              b0.y = sXc[(k0 + koff + 1) * XC_S + nn];
      v2f b1; b1.x = sXc[(k0 + koff) * XC_S + 16 + nn];
              b1.y = sXc[(k0 + koff + 1) * XC_S + 16 + nn];
      acc0 = __builtin_amdgcn_wmma_f32_16x16x4_f32(false, a, false, b0, (short)0, acc0, false, false);
      acc1 = __builtin_amdgcn_wmma_f32_16x16x4_f32(false, a, false, b1, (short)0, acc1, false, false);
    }
    float* sW = sY;                        // reuse solve scratch as W[m][b]
    const int mbase = m0 + ((lane >> 4) << 3);   // lanes 16-31 hold M = 8 + r
    for (int r = 0; r < 8; ++r) {
      sW[(mbase + r) * B_ + nn]      = acc0[r];
      sW[(mbase + r) * B_ + 16 + nn] = acc1[r];
    }
  }
  __syncthreads();

  // ---- Phase 3: cosine similarity per (b, p).
  {
    const int b   = tid >> 3;   // 0..31
    const int seg = tid & 7;    // 16 rows each
    float dot = 0.f, nw = 0.f;
    const float* sW = sY;
    for (int m = seg * 16; m < seg * 16 + 16; ++m) {
      float w;
      if (ctype == 0)       w = sW[m * B_ + b];
      else if (ctype == 1)  w = sXc[m * XC_S + b] * sDiag[m];  // sDiag = 1/diag(inv)
      else                  w = sXc[m * XC_S + b];   // isotropic scale cancels in L2 norm
      const float s = sSig[m];
      dot += w * s;
      nw  += w * w;
    }
    sRed[(b * 8 + seg) * 2 + 0] = dot;
    sRed[(b * 8 + seg) * 2 + 1] = nw;
  }
  __syncthreads();
  if ((tid & 7) == 0) {
    const int b = tid >> 3;
    float Dt = 0.f, Nw = 0.f;
    for (int q = 0; q < 8; ++q) {
      Dt += sRed[(b * 8 + q) * 2 + 0];
      Nw += sRed[(b * 8 + q) * 2 + 1];
    }
    out[(size_t)b * P_ + p] = Dt / (fmaxf(sqrtf(Nw), 1e-12f) * sNs[0]);
  }
}

extern "C" void kernel_launch(void* const* d_in, const int* in_sizes, int n_in,
                              void* d_out, int out_size, void* d_ws, size_t ws_size,
                              hipStream_t stream) {
  const float* X          = (const float*)d_in[0];
  const float* b_mean     = (const float*)d_in[1];
  const float* b_covs     = (const float*)d_in[2];
  const float* signatures = (const float*)d_in[3];
  const int*   cov_type   = (const int*)d_in[4];
  float* out = (float*)d_out;

  const size_t smem = (size_t)(D_ * COV_S + D_ * D_ + D_ * XC_S + D_ + D_ + D_ + 32 * 8 * 2 + 16)
                      * sizeof(float);  // ~152 KB; CDNA5 WGP has 320 KB LDS (2 WGs/WGP)
  hipFuncSetAttribute((const void*)ace_kernel,
                      hipFuncAttributeMaxDynamicSharedMemorySize, (int)smem);
  ace_kernel<<<P_, 256, smem, stream>>>(X, b_mean, b_covs, signatures, cov_type, out);
}